// FlashAttentionVarlen_63909113364576
// MI455X (gfx1250) — compile-verified
//
#include <hip/hip_runtime.h>
#include <hip/hip_bf16.h>

// CDNA5 (gfx1250) varlen flash-attention fwd, fp32 via V_WMMA_F32_16X16X4_F32.
// One wave (32 lanes) per block computes a 16(q) x 128(d) output tile for one
// (sequence, head). S is computed transposed (S^T = K * Q^T) so the softmax
// reduction axis maps to VGPRs + one SWAPX16 ds_swizzle, and PV is computed as
// O^T = V^T * P^T so P^T feeds the WMMA B operand straight from accumulators.

typedef __attribute__((ext_vector_type(2))) float v2f;
typedef __attribute__((ext_vector_type(8))) float v8f;

#define ATTN_H 16
#define ATTN_D 128
#define NEG_BIG (-3.0e38f)

__device__ __forceinline__ float swap16(float x) {
    // ds_swizzle SWAPX16: xor_mask=0x10, or=0, and=0x1f -> offset 0x401F
    return __builtin_bit_cast(float,
        __builtin_amdgcn_ds_swizzle(__builtin_bit_cast(int, x), 0x401F));
}

__global__ __launch_bounds__(32)
void fa_varlen_wmma(const float* __restrict__ Q,
                    const float* __restrict__ K,
                    const float* __restrict__ V,
                    const int* __restrict__ cu_q,
                    const int* __restrict__ cu_k,
                    float* __restrict__ O) {
    const int tile = blockIdx.x;
    const int h    = blockIdx.y;
    const int b    = blockIdx.z;

    const int q_begin = cu_q[b];
    const int q_endg  = cu_q[b + 1];
    const int k_begin = cu_k[b];
    const int k_endg  = cu_k[b + 1];
    const int Lq = q_endg - q_begin;
    const int Lk = k_endg - k_begin;
    if (tile * 16 >= Lq) return;          // uniform branch: EXEC stays full

    const int lane = threadIdx.x;         // wave32
    const int lm   = lane & 15;
    const int half = lane >> 4;           // selects upper K-pair of WMMA operand
    const bool lowHalf = (half == 0);

    const int  qrow_rel = tile * 16 + lm; // q column owned by this lane
    const bool qvalid   = qrow_rel < Lq;
    const int  qrow     = q_begin + (qvalid ? qrow_rel : (Lq - 1));

    const float scale = 0.08838834764831845f; // 1/sqrt(128), folded into Q

    // ---- Preload Q tile as B-operand slices of Q^T (resident all loop) ----
    // slice i (d0 = 4*i): lane holds Q[qrow][d0 + 2*half + {0,1}]
    v2f qreg[32];
    const float* qbase = Q + (size_t)qrow * (ATTN_H * ATTN_D) + h * ATTN_D + 2 * half;
    #pragma unroll
    for (int i = 0; i < 32; ++i) {
        v2f t = *(const v2f*)(qbase + 4 * i);
        t.x *= scale; t.y *= scale;
        qreg[i] = t;
    }

    v8f o[8];                             // O^T: 8 tiles of 16(d) x 16(q)
    #pragma unroll
    for (int t = 0; t < 8; ++t) o[t] = {};

    float m_run = NEG_BIG;
    float l_run = 0.0f;

    for (int kvb = 0; kvb < Lk; kvb += 16) {
        // ---- S^T = K_chunk * Q^T : A = K slice, B = Q^T slice ----
        const int krow_rel = kvb + lm;
        const int krow = k_begin + (krow_rel < Lk ? krow_rel : (Lk - 1));
        const float* kbase = K + (size_t)krow * (ATTN_H * ATTN_D) + h * ATTN_D + 2 * half;

        v8f sa = {}, sb = {};             // two chains to hide WMMA latency
        #pragma unroll
        for (int i = 0; i < 32; i += 2) {
            v2f a0 = *(const v2f*)(kbase + 4 * i);
            v2f a1 = *(const v2f*)(kbase + 4 * (i + 1));
            sa = __builtin_amdgcn_wmma_f32_16x16x4_f32(false, a0, false, qreg[i],
                                                       (short)0, sa, false, false);
            sb = __builtin_amdgcn_wmma_f32_16x16x4_f32(false, a1, false, qreg[i + 1],
                                                       (short)0, sb, false, false);
        }
        v8f s = sa + sb;

        // mask kv rows past the end: element j sits at kv row kvb + j + 8*half
        #pragma unroll
        for (int j = 0; j < 8; ++j)
            s[j] = ((kvb + j + 8 * half) < Lk) ? s[j] : NEG_BIG;

        // ---- online softmax: per-q (= per-lane) max/sum over kv ----
        float mloc = s[0];
        #pragma unroll
        for (int j = 1; j < 8; ++j) mloc = fmaxf(mloc, s[j]);
        mloc = fmaxf(mloc, swap16(mloc));               // combine halves
        const float m_new = fmaxf(m_run, mloc);
        const float alpha = __expf(m_run - m_new);      // 0 on first chunk

        float lloc = 0.0f;
        #pragma unroll
        for (int j = 0; j < 8; ++j) {
            float e = __expf(s[j] - m_new);
            s[j] = e;                                   // s now holds P^T
            lloc += e;
        }
        lloc += swap16(lloc);
        l_run = l_run * alpha + lloc;
        m_run = m_new;

        #pragma unroll
        for (int t = 0; t < 8; ++t) {
            #pragma unroll
            for (int j = 0; j < 8; ++j) o[t][j] *= alpha;
        }

        // cross-half image of P^T for B-operand assembly
        float ps[8];
        #pragma unroll
        for (int j = 0; j < 8; ++j) ps[j] = swap16(s[j]);

        // B-operand (P^T 4x16 slice) per k-slice sI:
        //  lanes 0-15 need rows {4sI,4sI+1}; lanes 16-31 need rows {4sI+2,4sI+3}
        const float b0h0[4] = { s[0],  s[4],  ps[0], ps[4] };
        const float b1h0[4] = { s[1],  s[5],  ps[1], ps[5] };
        const float b0h1[4] = { ps[2], ps[6], s[2],  s[6]  };
        const float b1h1[4] = { ps[3], ps[7], s[3],  s[7]  };

        // ---- O^T += V^T * P^T : A = V^T d-tile slice, B = P^T slice ----
        #pragma unroll
        for (int sI = 0; sI < 4; ++sI) {
            const int kk = 4 * sI + 2 * half;
            const int r0 = k_begin + ((kvb + kk)     < Lk ? (kvb + kk)     : (Lk - 1));
            const int r1 = k_begin + ((kvb + kk + 1) < Lk ? (kvb + kk + 1) : (Lk - 1));
            const float* v0p = V + (size_t)r0 * (ATTN_H * ATTN_D) + h * ATTN_D + lm;
            const float* v1p = V + (size_t)r1 * (ATTN_H * ATTN_D) + h * ATTN_D + lm;
            v2f bop;
            bop.x = lowHalf ? b0h0[sI] : b0h1[sI];
            bop.y = lowHalf ? b1h0[sI] : b1h1[sI];
            #pragma unroll
            for (int t = 0; t < 8; ++t) {
                v2f a;                     // A[m][kk..kk+1] = V[r][t*16+lm]
                a.x = v0p[t * 16];
                a.y = v1p[t * 16];
                o[t] = __builtin_amdgcn_wmma_f32_16x16x4_f32(false, a, false, bop,
                                                             (short)0, o[t], false, false);
            }
        }
    }

    // ---- epilogue: O[q][d] = O^T[d][q] / l ----
    if (qvalid) {
        const float inv_l = 1.0f / l_run;
        float* obase = O + (size_t)qrow * (ATTN_H * ATTN_D) + h * ATTN_D + 8 * half;
        #pragma unroll
        for (int t = 0; t < 8; ++t) {
            #pragma unroll
            for (int j = 0; j < 8; ++j)
                obase[t * 16 + j] = o[t][j] * inv_l;
        }
    }
}

extern "C" void kernel_launch(void* const* d_in, const int* in_sizes, int n_in,
                              void* d_out, int out_size, void* d_ws, size_t ws_size,
                              hipStream_t stream) {
    const float* q    = (const float*)d_in[0];
    const float* k    = (const float*)d_in[1];
    const float* v    = (const float*)d_in[2];
    const int*   cu_q = (const int*)d_in[3];
    const int*   cu_k = (const int*)d_in[4];
    float* out = (float*)d_out;

    const int B = in_sizes[3] - 1;                    // cu_seqlens has B+1 entries
    const int T = in_sizes[0] / (ATTN_H * ATTN_D);    // total packed tokens
    const int maxTiles = (T + 15) / 16;               // covers any seqlen <= T

    dim3 grid(maxTiles, ATTN_H, B);
    dim3 block(32, 1, 1);                             // one wave32 per block
    hipLaunchKernelGGL(fa_varlen_wmma, grid, block, 0, stream,
                       q, k, v, cu_q, cu_k, out);
}